// R_GCN_53197464928388
// MI455X (gfx1250) — compile-verified
//
#include <hip/hip_runtime.h>
#include <hip/hip_bf16.h>
#include <math.h>
#include <stdint.h>

// R-GCN (3 layers) for MI455X / gfx1250, fp32 end-to-end.
//   N=50000 nodes, E=800000 edges, R=8 relations, dims 128->256->128->128.
// Pipeline per layer:
//   1) scatter: agg[dst,rel,:] += inv_cnt[dst,rel] * h[src,:]   (f32 atomics;
//      the per-(dst,rel) scale is constant inside a segment, so folding it
//      into the scatter makes agg the exact per-relation *mean*)
//   2) grouped GEMM  out = ELU( agg_flat @ W_flat + h @ root + b ) with
//      V_WMMA_F32_16X16X4_F32, A-panels double-buffered into LDS through the
//      CDNA5 async-to-LDS path (ASYNCcnt, inline asm), B chunks preloaded
//      into registers.
// h1,h2 live inside d_out (concat layout, row stride 512) -> no extra copies.
// Workspace (floats): [ inv_cnt : N*R ][ agg : N*R*256 ]  (~411 MB).

#define N_NODES 50000
#define N_EDGES 800000
#define R_REL   8
#define F_IN    128
#define D1      256
#define D2      128
#define D3      128
#define OUT_STRIDE (D1 + D2 + D3)   // 512

typedef __attribute__((ext_vector_type(2))) float v2f;
typedef __attribute__((ext_vector_type(8))) float v8f;

// ---------------- CDNA5 async global->LDS copy (16B per lane) ----------------
// GLOBAL_LOAD_ASYNC_TO_LDS_B128: VDST VGPR carries the LDS byte address,
// VADDR pair carries the 64-bit global address. Tracked by ASYNCcnt.
// Generic pointers to __shared__ carry the LDS offset in their low 32 bits
// (shared aperture lives in addr[63:32]), so truncation yields the LDS addr.
__device__ __forceinline__ void async_cp16(const float* g, float* l) {
    unsigned lds_addr = (unsigned)(uintptr_t)l;
    unsigned long long gaddr = (unsigned long long)(uintptr_t)g;
    asm volatile("global_load_async_to_lds_b128 %0, %1, off"
                 :: "v"(lds_addr), "v"(gaddr)
                 : "memory");
}

__device__ __forceinline__ void wait_async() {
    asm volatile("s_wait_asynccnt 0x0" ::: "memory");
}

// ---------------------------------- utility ----------------------------------
__global__ void zero_kernel(float* __restrict__ p, long n) {
    long i = (long)blockIdx.x * blockDim.x + threadIdx.x;
    long stride = (long)gridDim.x * blockDim.x;
    for (; i < n; i += stride) p[i] = 0.0f;
}

__global__ void count_kernel(const int* __restrict__ dst,
                             const int* __restrict__ etype,
                             float* __restrict__ cnt) {
    int e = blockIdx.x * blockDim.x + threadIdx.x;
    if (e >= N_EDGES) return;
    atomicAdd(&cnt[dst[e] * R_REL + etype[e]], 1.0f);
}

__global__ void invert_kernel(float* __restrict__ cnt) {
    int i = blockIdx.x * blockDim.x + threadIdx.x;
    if (i >= N_NODES * R_REL) return;
    cnt[i] = 1.0f / fmaxf(cnt[i], 1.0f);
}

// agg[(dst*R+rel)*FIN + f..f+3] += invc[dst*R+rel] * h[src*h_stride + f..f+3]
template <int FIN>
__global__ void scatter_kernel(const float* __restrict__ h, int h_stride,
                               const int* __restrict__ src,
                               const int* __restrict__ dst,
                               const int* __restrict__ etype,
                               const float* __restrict__ invc,
                               float* __restrict__ agg) {
    constexpr int F4 = FIN / 4;
    long idx = (long)blockIdx.x * blockDim.x + threadIdx.x;
    if (idx >= (long)N_EDGES * F4) return;
    int e = (int)(idx / F4);
    int f = (int)(idx % F4) * 4;
    int seg = dst[e] * R_REL + etype[e];
    float s = invc[seg];
    const float4 v = *(const float4*)(h + (long)src[e] * h_stride + f);
    float* p = agg + (long)seg * FIN + f;
    atomicAdd(p + 0, v.x * s);
    atomicAdd(p + 1, v.y * s);
    atomicAdd(p + 2, v.z * s);
    atomicAdd(p + 3, v.w * s);
}

// ------------------------------- grouped GEMM --------------------------------
// One block = 16 nodes x 128 output cols (8 wave32s, one 16x16 D tile each).
// K = R*FIN (B = flat W) then FIN (B = root), chunked by 64, A double-buffered
// in LDS via async-to-LDS, B chunk preloaded into 32 registers per lane.
template <int FIN, int FOUT>
__global__ __launch_bounds__(256) void rgcn_gemm_kernel(
    const float* __restrict__ agg,   // [N, R*FIN] per-relation means
    const float* __restrict__ hin,   // [N, hin_stride]
    int hin_stride,
    const float* __restrict__ W,     // [R*FIN, FOUT] (flat [R,FIN,FOUT])
    const float* __restrict__ root,  // [FIN, FOUT]
    const float* __restrict__ bias,  // [FOUT]
    float* __restrict__ out, int out_stride)
{
    constexpr int KA  = R_REL * FIN;     // relation part of K
    constexpr int KT  = KA + FIN;        // total K
    constexpr int NCH = KT / 64;         // 64-wide K chunks
    constexpr int LD  = 68;              // LDS row stride: 16B-aligned, and
                                         // banks (4*row+c)%64 -> conflict-free
    __shared__ __align__(16) float As[2][16][LD];

    const int tid   = threadIdx.x;
    const int wave  = tid >> 5;
    const int lane  = tid & 31;
    const int row   = lane & 15;           // A row / D column inside tile
    const int khalf = (lane >> 4) << 1;    // K-pair select (ISA 16x4x16 layout)
    const int nodeBase = blockIdx.x * 16;
    const int col = blockIdx.y * 128 + wave * 16 + row;

    // staging slice: each thread async-copies 16B of the A panel
    const int rowA = tid >> 4;             // 0..15
    const int kkA  = (tid & 15) * 4;       // 0,4,...,60
    const float* aggRow = agg + (long)(nodeBase + rowA) * KA + kkA;
    const float* hinRow = hin + (long)(nodeBase + rowA) * hin_stride + kkA;

    auto stage = [&](int c, int buf) {
        const int kc = c * 64;
        const float* g = (kc < KA) ? (aggRow + kc) : (hinRow + (kc - KA));
        async_cp16(g, &As[buf][rowA][kkA]);
    };

    v8f acc = {};
    stage(0, 0);

    for (int c = 0; c < NCH; ++c) {
        const int buf = c & 1;
        wait_async();          // this wave's chunk c has landed in LDS
        __syncthreads();       // ... and every other wave's slice too;
                               // also: everyone finished reading buf^1
        if (c + 1 < NCH) stage(c + 1, buf ^ 1);   // overlap with compute

        const int kc = c * 64;
        const float* Bp = (kc < KA) ? (W + (long)kc * FOUT + col)
                                    : (root + (long)(kc - KA) * FOUT + col);
        __builtin_prefetch(Bp + 64 * FOUT, 0, 3);

        float br[32];                          // whole B chunk -> registers
        #pragma unroll
        for (int s = 0; s < 16; ++s) {
            br[2 * s]     = Bp[(4 * s + khalf) * FOUT];
            br[2 * s + 1] = Bp[(4 * s + khalf + 1) * FOUT];
        }

        const float* arow = &As[buf][row][0];
        #pragma unroll
        for (int s = 0; s < 16; ++s) {
            v2f a, b;
            a.x = arow[4 * s + khalf];
            a.y = arow[4 * s + khalf + 1];
            b.x = br[2 * s];
            b.y = br[2 * s + 1];
            acc = __builtin_amdgcn_wmma_f32_16x16x4_f32(
                false, a, false, b, (short)0, acc, false, false);
        }
        // no trailing barrier needed: next iteration's wait+barrier protects
        // both the new data and the buffer being overwritten after that
    }

    // ---- epilogue: bias + ELU, D scatter per ISA 16x16 f32 C/D layout ----
    const float bval = bias[col];
    const int mAdd = (lane >> 4) * 8;          // lanes 16-31 hold rows M+8
    #pragma unroll
    for (int g = 0; g < 8; ++g) {
        float v = acc[g] + bval;
        v = v > 0.0f ? v : expm1f(v);          // ELU(alpha=1)
        out[(long)(nodeBase + g + mAdd) * out_stride + col] = v;
    }
}

// --------------------------------- launcher ----------------------------------
extern "C" void kernel_launch(void* const* d_in, const int* in_sizes, int n_in,
                              void* d_out, int out_size, void* d_ws, size_t ws_size,
                              hipStream_t stream) {
    const float* x     = (const float*)d_in[0];
    const int*   eidx  = (const int*)d_in[1];   // [2, E]
    const int*   etype = (const int*)d_in[2];
    const float* w1    = (const float*)d_in[3];
    const float* root1 = (const float*)d_in[4];
    const float* b1    = (const float*)d_in[5];
    const float* w2    = (const float*)d_in[6];
    const float* root2 = (const float*)d_in[7];
    const float* b2    = (const float*)d_in[8];
    const float* w3    = (const float*)d_in[9];
    const float* root3 = (const float*)d_in[10];
    const float* b3    = (const float*)d_in[11];
    float* out = (float*)d_out;

    const int* src = eidx;
    const int* dst = eidx + N_EDGES;

    float* invc = (float*)d_ws;                        // N*R
    float* agg  = invc + (size_t)N_NODES * R_REL;      // N*R*256 (reused)

    // degree counts -> inverse (layer independent, needed before scatters)
    zero_kernel<<<1024, 256, 0, stream>>>(invc, (long)N_NODES * R_REL);
    count_kernel<<<(N_EDGES + 255) / 256, 256, 0, stream>>>(dst, etype, invc);
    invert_kernel<<<(N_NODES * R_REL + 255) / 256, 256, 0, stream>>>(invc);

    // ---- Layer 1: 128 -> 256, h1 = out[:, 0:256] ----
    {
        zero_kernel<<<4096, 256, 0, stream>>>(agg, (long)N_NODES * R_REL * F_IN);
        long th = (long)N_EDGES * (F_IN / 4);
        scatter_kernel<F_IN><<<(th + 255) / 256, 256, 0, stream>>>(
            x, F_IN, src, dst, etype, invc, agg);
        dim3 g(N_NODES / 16, D1 / 128);
        rgcn_gemm_kernel<F_IN, D1><<<g, 256, 0, stream>>>(
            agg, x, F_IN, w1, root1, b1, out, OUT_STRIDE);
    }

    // ---- Layer 2: 256 -> 128, h2 = out[:, 256:384] ----
    {
        zero_kernel<<<4096, 256, 0, stream>>>(agg, (long)N_NODES * R_REL * D1);
        long th = (long)N_EDGES * (D1 / 4);
        scatter_kernel<D1><<<(th + 255) / 256, 256, 0, stream>>>(
            out, OUT_STRIDE, src, dst, etype, invc, agg);
        dim3 g(N_NODES / 16, D2 / 128);
        rgcn_gemm_kernel<D1, D2><<<g, 256, 0, stream>>>(
            agg, out, OUT_STRIDE, w2, root2, b2, out + D1, OUT_STRIDE);
    }

    // ---- Layer 3: 128 -> 128, h3 = out[:, 384:512] ----
    {
        zero_kernel<<<4096, 256, 0, stream>>>(agg, (long)N_NODES * R_REL * D2);
        long th = (long)N_EDGES * (D2 / 4);
        scatter_kernel<D2><<<(th + 255) / 256, 256, 0, stream>>>(
            out + D1, OUT_STRIDE, src, dst, etype, invc, agg);
        dim3 g(N_NODES / 16, D3 / 128);
        rgcn_gemm_kernel<D2, D3><<<g, 256, 0, stream>>>(
            agg, out + D1, OUT_STRIDE, w3, root3, b3,
            out + D1 + D2, OUT_STRIDE);
    }
}